// NeuralIF_80470507258346
// MI455X (gfx1250) — compile-verified
//
#include <hip/hip_runtime.h>
#include <math.h>

typedef __attribute__((ext_vector_type(2))) float v2f;
typedef __attribute__((ext_vector_type(8))) float v8f;

static __device__ __forceinline__ v8f wmma4(v2f a, v2f b, v8f c) {
    // D = A(16x4 f32) * B(4x16 f32) + C(16x16 f32), exact f32 via matrix unit
    return __builtin_amdgcn_wmma_f32_16x16x4_f32(false, a, false, b, (short)0, c,
                                                 false, false);
}

// ---------------- Edge MLP: concat([g, x[row], x[col], e_prev, skip]) -> 32 -> 1
// One wave handles 16 edges. 256 threads = 8 waves = 128 edges per block.
__global__ __launch_bounds__(256) void edge_kernel(
    const float* __restrict__ xs,    // N x 8
    const int*   __restrict__ row,   // E
    const int*   __restrict__ col,   // E
    const float* __restrict__ eprev, // E
    const float* __restrict__ skip,  // E or nullptr
    const float* __restrict__ gg,    // 8
    const float* __restrict__ W1,    // 26 x 32 row-major
    const float* __restrict__ B1,    // 32
    const float* __restrict__ W2,    // 32 x 1
    const float* __restrict__ B2,    // 1
    float* __restrict__ enew,        // E
    float* __restrict__ agg,         // N (atomic)
    float* __restrict__ cnt,         // N (atomic)
    float* __restrict__ esum,        // 1 (atomic)
    int E)
{
    __shared__ float hid[8][16][33];
    __shared__ float bsum;
    const int tid  = threadIdx.x;
    const int w    = tid >> 5;
    const int lane = tid & 31;
    const int m    = lane & 15;   // edge within tile / N index
    const int h    = lane >> 4;   // half selector
    const int n    = m;
    const int base = (blockIdx.x * 8 + w) * 16;

    if (tid == 0) bsum = 0.f;

    const int eM = base + m;
    const int eC = eM < E ? eM : (E - 1);
    const int r  = row[eC];
    const int c  = col[eC];

    v2f g0  = { gg[2*h],   gg[2*h+1] };
    v2f g1  = { gg[4+2*h], gg[5+2*h] };
    v2f xr0 = *(const v2f*)(xs + (size_t)r*8 + 2*h);
    v2f xr1 = *(const v2f*)(xs + (size_t)r*8 + 4 + 2*h);
    v2f xc0 = *(const v2f*)(xs + (size_t)c*8 + 2*h);
    v2f xc1 = *(const v2f*)(xs + (size_t)c*8 + 4 + 2*h);

    // unconditional loads + lane select (no exec-mask branches)
    const float ep = eprev[eC];
    const float sk = (skip != nullptr) ? skip[eC] : 0.f;   // uniform pointer test
    v2f es;
    es.x = (h == 0) ? ep : 0.f;
    es.y = (h == 0) ? sk : 0.f;

    v2f afrag[7] = { g0, g1, xr0, xr1, xc0, xc1, es };

    v8f acc0, acc1;
    {
        const float bb0 = B1[n], bb1 = B1[16 + n];
        #pragma unroll
        for (int i = 0; i < 8; ++i) { acc0[i] = bb0; acc1[i] = bb1; }
    }
    #pragma unroll
    for (int kc = 0; kc < 7; ++kc) {
        const int f0 = 4*kc + 2*h;
        const int fa = (f0     < 26) ? f0     : 0;   // clamped addresses, always safe
        const int fb = (f0 + 1 < 26) ? f0 + 1 : 0;
        const bool va = (f0 < 26), vb = (f0 + 1 < 26);
        const float w0a = W1[(size_t)fa*32 + n];
        const float w0b = W1[(size_t)fb*32 + n];
        const float w1a = W1[(size_t)fa*32 + 16 + n];
        const float w1b = W1[(size_t)fb*32 + 16 + n];
        v2f bw0, bw1;
        bw0.x = va ? w0a : 0.f;  bw0.y = vb ? w0b : 0.f;
        bw1.x = va ? w1a : 0.f;  bw1.y = vb ? w1b : 0.f;
        acc0 = wmma4(afrag[kc], bw0, acc0);
        acc1 = wmma4(afrag[kc], bw1, acc1);
    }
    #pragma unroll
    for (int i = 0; i < 8; ++i) {        // ReLU, spill hidden to LDS
        hid[w][i + 8*h][n]      = fmaxf(acc0[i], 0.f);
        hid[w][i + 8*h][16 + n] = fmaxf(acc1[i], 0.f);
    }
    __syncthreads();

    v8f acc2;
    {
        const float b2v = B2[0];                     // uniform load
        const float bb  = (n == 0) ? b2v : 0.f;
        #pragma unroll
        for (int i = 0; i < 8; ++i) acc2[i] = bb;
    }
    #pragma unroll
    for (int kc = 0; kc < 8; ++kc) {
        const int f0 = 4*kc + 2*h;
        v2f a2;  a2.x = hid[w][m][f0];  a2.y = hid[w][m][f0+1];
        const float wva = W2[f0];                    // uniform only within each lane half
        const float wvb = W2[f0+1];
        v2f b2f;
        b2f.x = (n == 0) ? wva : 0.f;
        b2f.y = (n == 0) ? wvb : 0.f;
        acc2 = wmma4(a2, b2f, acc2);
    }

    float myEsum = 0.f;
    if (n == 0) {
        #pragma unroll
        for (int rr = 0; rr < 8; ++rr) {
            const int e2 = base + rr + 8*h;
            if (e2 < E) {
                const float v = acc2[rr];
                enew[e2] = v;
                const int ri = row[e2];
                atomicAdd(agg + ri, v);
                atomicAdd(cnt + ri, 1.0f);
                myEsum += v;
            }
        }
        atomicAdd(&bsum, myEsum);
    }
    __syncthreads();
    if (tid == 0) atomicAdd(esum, bsum);
}

// ---------------- Node MLP: concat([g, x, agg]) -> 32 -> 8
__global__ __launch_bounds__(256) void node_kernel(
    const float* __restrict__ xs,   // N x 8
    const float* __restrict__ agg,  // N
    const float* __restrict__ cnt,  // N
    const float* __restrict__ gg,   // 8
    const float* __restrict__ W1,   // 17 x 32
    const float* __restrict__ B1,   // 32
    const float* __restrict__ W2,   // 32 x 8
    const float* __restrict__ B2,   // 8
    float* __restrict__ nout,       // N x 8 or nullptr
    float* __restrict__ nsum,       // 8 (atomic)
    int N)
{
    __shared__ float hid[8][16][33];
    __shared__ float csum[8];
    const int tid  = threadIdx.x;
    const int w    = tid >> 5;
    const int lane = tid & 31;
    const int m    = lane & 15;
    const int h    = lane >> 4;
    const int n    = m;
    const int base = (blockIdx.x * 8 + w) * 16;

    if (tid < 8) csum[tid] = 0.f;

    const int iM = base + m;
    const int iC = iM < N ? iM : (N - 1);

    v2f g0 = { gg[2*h],   gg[2*h+1] };
    v2f g1 = { gg[4+2*h], gg[5+2*h] };
    v2f x0 = *(const v2f*)(xs + (size_t)iC*8 + 2*h);
    v2f x1 = *(const v2f*)(xs + (size_t)iC*8 + 4 + 2*h);

    // unconditional loads + select (no exec-mask branch)
    const float cc = cnt[iC];
    const float av = agg[iC] / fmaxf(cc, 1.0f);
    v2f a4;
    a4.x = (h == 0) ? av : 0.f;
    a4.y = 0.f;

    v2f afrag[5] = { g0, g1, x0, x1, a4 };

    v8f acc0, acc1;
    {
        const float bb0 = B1[n], bb1 = B1[16 + n];
        #pragma unroll
        for (int i = 0; i < 8; ++i) { acc0[i] = bb0; acc1[i] = bb1; }
    }
    #pragma unroll
    for (int kc = 0; kc < 5; ++kc) {
        const int f0 = 4*kc + 2*h;
        const int fa = (f0     < 17) ? f0     : 0;
        const int fb = (f0 + 1 < 17) ? f0 + 1 : 0;
        const bool va = (f0 < 17), vb = (f0 + 1 < 17);
        const float w0a = W1[(size_t)fa*32 + n];
        const float w0b = W1[(size_t)fb*32 + n];
        const float w1a = W1[(size_t)fa*32 + 16 + n];
        const float w1b = W1[(size_t)fb*32 + 16 + n];
        v2f bw0, bw1;
        bw0.x = va ? w0a : 0.f;  bw0.y = vb ? w0b : 0.f;
        bw1.x = va ? w1a : 0.f;  bw1.y = vb ? w1b : 0.f;
        acc0 = wmma4(afrag[kc], bw0, acc0);
        acc1 = wmma4(afrag[kc], bw1, acc1);
    }
    #pragma unroll
    for (int i = 0; i < 8; ++i) {
        hid[w][i + 8*h][n]      = fmaxf(acc0[i], 0.f);
        hid[w][i + 8*h][16 + n] = fmaxf(acc1[i], 0.f);
    }
    __syncthreads();

    const int nc = n & 7;                 // clamped column for safe loads
    v8f acc2;
    {
        const float bv = B2[nc];
        const float bb = (n < 8) ? bv : 0.f;
        #pragma unroll
        for (int i = 0; i < 8; ++i) acc2[i] = bb;
    }
    #pragma unroll
    for (int kc = 0; kc < 8; ++kc) {
        const int f0 = 4*kc + 2*h;
        v2f a2;  a2.x = hid[w][m][f0];  a2.y = hid[w][m][f0+1];
        const float wva = W2[(size_t)f0*8 + nc];
        const float wvb = W2[(size_t)(f0+1)*8 + nc];
        v2f b2f;
        b2f.x = (n < 8) ? wva : 0.f;
        b2f.y = (n < 8) ? wvb : 0.f;
        acc2 = wmma4(a2, b2f, acc2);
    }

    if (n < 8) {
        float colsum = 0.f;
        #pragma unroll
        for (int rr = 0; rr < 8; ++rr) {
            const int i2 = base + rr + 8*h;
            if (i2 < N) {
                const float v = acc2[rr];
                if (nout != nullptr) nout[(size_t)i2*8 + n] = v;
                colsum += v;
            }
        }
        atomicAdd(&csum[n], colsum);
    }
    __syncthreads();
    if (tid < 8) atomicAdd(nsum + tid, csum[tid]);
}

// ---------------- Global MLP: concat([n_mean, e_mean, g]) -> 32 -> 8 ; updates g in place
__global__ __launch_bounds__(32) void global_kernel(
    const float* __restrict__ nsum, const float* __restrict__ esum,
    float* __restrict__ g,
    const float* __restrict__ W1,  // 17 x 32
    const float* __restrict__ B1,  // 32
    const float* __restrict__ W2,  // 32 x 8
    const float* __restrict__ B2,  // 8
    float invN, float invE)
{
    __shared__ float u[17];
    __shared__ float hsh[32];
    const int t = threadIdx.x;
    if (t < 17) u[t] = (t < 8) ? nsum[t] * invN : ((t == 8) ? esum[0] * invE : g[t - 9]);
    __syncthreads();
    float hh = B1[t];
    #pragma unroll
    for (int k = 0; k < 17; ++k) hh += u[k] * W1[k*32 + t];
    hsh[t] = fmaxf(hh, 0.f);
    __syncthreads();
    if (t < 8) {
        float o = B2[t];
        #pragma unroll
        for (int j = 0; j < 32; ++j) o += hsh[j] * W2[j*8 + t];
        g[t] = o;
    }
}

// ---------------- Finalize: exp() on diagonal edges, concat lower|upper
__global__ __launch_bounds__(256) void final_kernel(
    const float* __restrict__ le, const float* __restrict__ ue,
    const int* __restrict__ lr, const int* __restrict__ lc,
    const int* __restrict__ ur, const int* __restrict__ uc,
    float* __restrict__ out, int E)
{
    const int j = blockIdx.x * blockDim.x + threadIdx.x;
    if (j < E) {
        const float v = le[j];
        out[j] = (lr[j] == lc[j]) ? expf(v) : v;
    } else if (j < 2 * E) {
        const int k = j - E;
        const float v = ue[k];
        out[j] = (ur[k] == uc[k]) ? expf(v) : v;
    }
}

extern "C" void kernel_launch(void* const* d_in, const int* in_sizes, int n_in,
                              void* d_out, int out_size, void* d_ws, size_t ws_size,
                              hipStream_t stream) {
    const float* x     = (const float*)d_in[0];
    const int*   lei   = (const int*)  d_in[1];
    const int*   uei   = (const int*)  d_in[2];
    const float* lattr = (const float*)d_in[3];
    const float* uattr = (const float*)d_in[4];
    const float* eW1 = (const float*)d_in[5];
    const float* eb1 = (const float*)d_in[6];
    const float* eW2 = (const float*)d_in[7];
    const float* eb2 = (const float*)d_in[8];
    const float* nW1 = (const float*)d_in[9];
    const float* nb1 = (const float*)d_in[10];
    const float* nW2 = (const float*)d_in[11];
    const float* nb2 = (const float*)d_in[12];
    const float* gW1 = (const float*)d_in[13];
    const float* gb1 = (const float*)d_in[14];
    const float* gW2 = (const float*)d_in[15];
    const float* gb2 = (const float*)d_in[16];

    const int N = in_sizes[0] / 8;
    const int E = in_sizes[3];
    const int *lr = lei, *lc = lei + E, *ur = uei, *uc = uei + E;

    // workspace carve (floats): l_e[E] u_e[E] l_n[N*8] agg[N] cnt[N] esum[1] nsum[8] g[8]
    float* ws   = (float*)d_ws;
    float* l_e  = ws;
    float* u_e  = l_e + E;
    float* l_n  = u_e + E;
    float* agg  = l_n + (size_t)N * 8;
    float* cnt  = agg + N;
    float* esum = cnt + N;
    float* nsum = esum + 1;
    float* g    = nsum + 8;

    hipMemsetAsync(g, 0, 8 * sizeof(float), stream);

    const int egrid = (E + 127) / 128;
    const int ngrid = (N + 127) / 128;
    const float invN = 1.0f / (float)N;
    const float invE = 1.0f / (float)E;

    for (int layer = 0; layer < 3; ++layer) {
        for (int b = 0; b < 2; ++b) {
            const int idx = b * 3 + layer;               // (2, N_LAYERS, ...) indexing
            const float* xs    = (b == 0) ? x : l_n;
            const int*   row   = (b == 0) ? lr : ur;
            const int*   col   = (b == 0) ? lc : uc;
            const float* eprev = (layer == 0) ? ((b == 0) ? lattr : uattr)
                                              : ((b == 0) ? l_e : u_e);
            const float* skip  = (b == 0 && layer > 0) ? lattr : nullptr;
            float* enew = (b == 0) ? l_e : u_e;
            float* nout = (b == 0) ? l_n : nullptr;      // upper n_emb only feeds the mean

            // zero agg, cnt, esum, nsum in one shot (contiguous)
            hipMemsetAsync(agg, 0, (size_t)(2 * N + 9) * sizeof(float), stream);

            edge_kernel<<<egrid, 256, 0, stream>>>(
                xs, row, col, eprev, skip, g,
                eW1 + (size_t)idx * 26 * 32, eb1 + idx * 32,
                eW2 + (size_t)idx * 32,      eb2 + idx,
                enew, agg, cnt, esum, E);

            node_kernel<<<ngrid, 256, 0, stream>>>(
                xs, agg, cnt, g,
                nW1 + (size_t)idx * 17 * 32, nb1 + idx * 32,
                nW2 + (size_t)idx * 32 * 8,  nb2 + idx * 8,
                nout, nsum, N);

            global_kernel<<<1, 32, 0, stream>>>(
                nsum, esum, g,
                gW1 + (size_t)idx * 17 * 32, gb1 + idx * 32,
                gW2 + (size_t)idx * 32 * 8,  gb2 + idx * 8,
                invN, invE);
        }
    }

    final_kernel<<<(2 * E + 255) / 256, 256, 0, stream>>>(
        l_e, u_e, lr, lc, ur, uc, (float*)d_out, E);
}